// GATLayer_13984413516292
// MI455X (gfx1250) — compile-verified
//
#include <hip/hip_runtime.h>
#include <math.h>

#define IN_DIM   128
#define K3       384          // 3 * IN_DIM
#define OUT_DIM  64
#define TILE_ROWS 64          // rows of z per block (4 waves x 16)
#define GEMM_THREADS 128      // 4 wave32s
#define WT_STRIDE 388         // padded column stride for transposed W (banks: 388%64=4)

typedef __attribute__((ext_vector_type(2))) float v2f;
typedef __attribute__((ext_vector_type(8))) float v8f;
typedef __attribute__((ext_vector_type(4))) int   v4i;

// gfx1250 async global->LDS path (ASYNCcnt-tracked), with sync fallback.
#if defined(__has_builtin)
#if __has_builtin(__builtin_amdgcn_global_load_async_to_lds_b128) && \
    __has_builtin(__builtin_amdgcn_s_wait_asynccnt)
#define HAVE_ASYNC_LDS 1
#endif
#endif

typedef __attribute__((address_space(1))) v4i glb_v4i;   // global int4
typedef __attribute__((address_space(3))) v4i lds_v4i;   // LDS int4

// ---------------------------------------------------------------------------
// Kernel 1: z = h3 @ W_fc via V_WMMA_F32_16X16X4_F32.
// h3[r] = [h[r-1], h[r], h[r+1]] is formed implicitly via LDS row offsets.
// LDS: W_fc transposed+padded (97 KB) + 66-row h slab with halo (33 KB).
// h slab is staged with GLOBAL_LOAD_ASYNC_TO_LDS_B128 when available.
// ---------------------------------------------------------------------------
__global__ __launch_bounds__(GEMM_THREADS) void gat_gemm_wmma(
    const float* __restrict__ h, const float* __restrict__ Wfc,
    float* __restrict__ z, int N)
{
    __shared__ float ldsWt[OUT_DIM * WT_STRIDE];       // 24832 f32, col-major W
    __shared__ float ldsH[(TILE_ROWS + 2) * IN_DIM];   //  8448 f32

    const int tid = threadIdx.x;
    const int R0  = blockIdx.x * TILE_ROWS;

    // Stage h rows [R0-1, R0+TILE_ROWS] (zero-padded halo) into LDS:
    // async b128 copies, one float4 per iteration per thread.
    {
        const int n4 = ((TILE_ROWS + 2) * IN_DIM) / 4;   // 2112 float4
        for (int i = tid; i < n4; i += GEMM_THREADS) {
            const int row  = R0 - 1 + (i >> 5);          // 32 float4 per row
            const int col4 = i & 31;
            float4* ldst = ((float4*)ldsH) + i;
            if (row >= 0 && row < N) {
#if defined(HAVE_ASYNC_LDS)
                __builtin_amdgcn_global_load_async_to_lds_b128(
                    (glb_v4i*)(h + (size_t)row * IN_DIM + col4 * 4),
                    (lds_v4i*)ldst, /*offset=*/0, /*cpol=*/0);
#else
                *ldst = ((const float4*)(h + (size_t)row * IN_DIM))[col4];
#endif
            } else {
                *ldst = make_float4(0.f, 0.f, 0.f, 0.f);
            }
        }
    }
    // Stage W_fc transposed: ldsWt[n*WT_STRIDE + k] = W[k][n].
    // Coalesced global reads (consecutive tid -> consecutive n of one row).
    for (int i = tid; i < K3 * OUT_DIM; i += GEMM_THREADS) {
        const int k = i >> 6;
        const int n = i & 63;
        ldsWt[n * WT_STRIDE + k] = Wfc[i];
    }
#if defined(HAVE_ASYNC_LDS)
    __builtin_amdgcn_s_wait_asynccnt(0);   // own wave's async copies done
#endif
    __syncthreads();                        // publish slab workgroup-wide

    const int wave = tid >> 5;        // wave32
    const int lane = tid & 31;
    const int r0   = wave * 16;       // 16-row tile inside the block
    const int m    = lane & 15;
    const int hi   = lane >> 4;       // lane group (0/1)

    v8f acc0 = {}; v8f acc1 = {}; v8f acc2 = {}; v8f acc3 = {};

    for (int kk = 0; kk < K3; kk += 4) {
        const int k0 = kk + 2 * hi;               // A/B: K = j + 2*(lane>>4)
        const int b  = k0 >> 7;                   // which 128-col block of h3
        const int ci = k0 & 127;
        // h3 row (R0+r0+m), block b -> h row (r+b-1) -> LDS row (r0+m+b)
        const v2f a = *(const v2f*)&ldsH[(r0 + m + b) * IN_DIM + ci];

        // Column-major W: {W[k0][n], W[k0+1][n]} is one aligned b64 load.
        const float* wb = &ldsWt[k0];
        const v2f b0 = *(const v2f*)&wb[( 0 + m) * WT_STRIDE];
        const v2f b1 = *(const v2f*)&wb[(16 + m) * WT_STRIDE];
        const v2f b2 = *(const v2f*)&wb[(32 + m) * WT_STRIDE];
        const v2f b3 = *(const v2f*)&wb[(48 + m) * WT_STRIDE];

        acc0 = __builtin_amdgcn_wmma_f32_16x16x4_f32(false, a, false, b0, (short)0, acc0, false, false);
        acc1 = __builtin_amdgcn_wmma_f32_16x16x4_f32(false, a, false, b1, (short)0, acc1, false, false);
        acc2 = __builtin_amdgcn_wmma_f32_16x16x4_f32(false, a, false, b2, (short)0, acc2, false, false);
        acc3 = __builtin_amdgcn_wmma_f32_16x16x4_f32(false, a, false, b3, (short)0, acc3, false, false);
    }

    // C/D layout: VGPR vv, lane -> row = vv + 8*hi, col = nt*16 + m.
    for (int vv = 0; vv < 8; ++vv) {
        const int row = R0 + r0 + vv + 8 * hi;
        if (row < N) {
            float* zr = z + (size_t)row * OUT_DIM;
            zr[ 0 + m] = acc0[vv];
            zr[16 + m] = acc1[vv];
            zr[32 + m] = acc2[vv];
            zr[48 + m] = acc3[vv];
        }
    }
}

// ---------------------------------------------------------------------------
// Kernel 2: per-node attention scalars a_src/a_dst = z[n] . Wa_{src,dst};
// also initializes emax=-inf, denom=0, out=0. One wave32 per node.
// ---------------------------------------------------------------------------
__global__ __launch_bounds__(128) void gat_node_scalars(
    const float* __restrict__ z, const float* __restrict__ W_attn,
    float* __restrict__ a_src, float* __restrict__ a_dst,
    float* __restrict__ emax, float* __restrict__ denom,
    float* __restrict__ out, int N)
{
    const int lane = threadIdx.x & 31;
    const int n    = blockIdx.x * 4 + (threadIdx.x >> 5);
    if (n >= N) return;

    const float* zr = z + (size_t)n * OUT_DIM;
    const float z0 = zr[lane], z1 = zr[lane + 32];
    float s = z0 * W_attn[lane]      + z1 * W_attn[lane + 32];
    float d = z0 * W_attn[64 + lane] + z1 * W_attn[96 + lane];
    for (int off = 16; off > 0; off >>= 1) {
        s += __shfl_xor(s, off);
        d += __shfl_xor(d, off);
    }
    out[(size_t)n * OUT_DIM + lane]      = 0.f;
    out[(size_t)n * OUT_DIM + lane + 32] = 0.f;
    if (lane == 0) {
        a_src[n] = s;
        a_dst[n] = d;
        emax[n]  = -INFINITY;
        denom[n] = 0.f;
    }
}

// Float atomic max via signed/unsigned integer atomics (order-safe trick).
__device__ __forceinline__ void atomicMaxF32(float* addr, float val) {
    if (val >= 0.f)
        atomicMax((int*)addr, __float_as_int(val));
    else
        atomicMin((unsigned int*)addr, (unsigned int)__float_as_int(val));
}

// ---------------------------------------------------------------------------
// Kernel 3: e[k] = a_src[src[k]] + a_dst[dst[k]]; segment max into emax.
// ---------------------------------------------------------------------------
__global__ __launch_bounds__(256) void gat_edge_logits(
    const float* __restrict__ a_src, const float* __restrict__ a_dst,
    const int* __restrict__ src, const int* __restrict__ dst,
    float* __restrict__ e, float* __restrict__ emax, int E)
{
    const int k = blockIdx.x * blockDim.x + threadIdx.x;
    if (k >= E) return;
    const float v = a_src[src[k]] + a_dst[dst[k]];
    e[k] = v;
    atomicMaxF32(&emax[dst[k]], v);
}

// ---------------------------------------------------------------------------
// Kernel 4: ex[k] = exp(e[k]-emax[dst]); denom[dst] += ex (in-place over e).
// ---------------------------------------------------------------------------
__global__ __launch_bounds__(256) void gat_edge_exp(
    const float* __restrict__ emax, const int* __restrict__ dst,
    float* __restrict__ e, float* __restrict__ denom, int E)
{
    const int k = blockIdx.x * blockDim.x + threadIdx.x;
    if (k >= E) return;
    const int d = dst[k];
    const float x = __expf(e[k] - emax[d]);
    e[k] = x;
    atomicAdd(&denom[d], x);
}

// ---------------------------------------------------------------------------
// Kernel 5: out[dst] += (ex/denom[dst]) * z[src]. One wave32 per edge,
// float2 per lane, native global_atomic_add_f32.
// ---------------------------------------------------------------------------
__global__ __launch_bounds__(256) void gat_aggregate(
    const float* __restrict__ z, const float* __restrict__ ex,
    const float* __restrict__ denom,
    const int* __restrict__ src, const int* __restrict__ dst,
    float* __restrict__ out, int E)
{
    const int widx = (int)((blockIdx.x * blockDim.x + threadIdx.x) >> 5);
    const int lane = threadIdx.x & 31;
    if (widx >= E) return;
    const int s = src[widx];
    const int d = dst[widx];
    const float alpha = ex[widx] / denom[d];
    const float2 v = ((const float2*)(z + (size_t)s * OUT_DIM))[lane];
    float* o = out + (size_t)d * OUT_DIM + lane * 2;
    atomicAdd(o + 0, alpha * v.x);
    atomicAdd(o + 1, alpha * v.y);
}

// ---------------------------------------------------------------------------
extern "C" void kernel_launch(void* const* d_in, const int* in_sizes, int n_in,
                              void* d_out, int out_size, void* d_ws, size_t ws_size,
                              hipStream_t stream)
{
    const float* h      = (const float*)d_in[0];
    const float* W_fc   = (const float*)d_in[1];
    const float* W_attn = (const float*)d_in[2];
    const int*   src    = (const int*)d_in[3];
    const int*   dst    = (const int*)d_in[4];
    float*       out    = (float*)d_out;

    const int N = in_sizes[0] / IN_DIM;
    const int E = in_sizes[3];

    // Workspace layout (256B aligned): z | a_src | a_dst | e/ex | emax | denom
    char* ws = (char*)d_ws;
    size_t off = 0;
    auto carve = [&](size_t bytes) {
        char* p = ws + off;
        off += (bytes + 255) & ~(size_t)255;
        return p;
    };
    float* z     = (float*)carve((size_t)N * OUT_DIM * sizeof(float));
    float* a_src = (float*)carve((size_t)N * sizeof(float));
    float* a_dst = (float*)carve((size_t)N * sizeof(float));
    float* e     = (float*)carve((size_t)E * sizeof(float));
    float* emax  = (float*)carve((size_t)N * sizeof(float));
    float* denom = (float*)carve((size_t)N * sizeof(float));
    (void)ws_size; (void)n_in; (void)out_size;

    // 1) GEMM (WMMA f32, async LDS staging)
    gat_gemm_wmma<<<(N + TILE_ROWS - 1) / TILE_ROWS, GEMM_THREADS, 0, stream>>>(
        h, W_fc, z, N);
    // 2) node scalars + init (emax, denom, out)
    gat_node_scalars<<<(N + 3) / 4, 128, 0, stream>>>(
        z, W_attn, a_src, a_dst, emax, denom, out, N);
    // 3) edge logits + segment max
    gat_edge_logits<<<(E + 255) / 256, 256, 0, stream>>>(
        a_src, a_dst, src, dst, e, emax, E);
    // 4) exp + segment sum
    gat_edge_exp<<<(E + 255) / 256, 256, 0, stream>>>(
        emax, dst, e, denom, E);
    // 5) weighted scatter aggregate (one wave per edge)
    gat_aggregate<<<(E * 8 + 255) / 256 + 1, 256, 0, stream>>>(
        z, e, denom, src, dst, out, E);
}